// CTCLayer_31731218383078
// MI455X (gfx1250) — compile-verified
//
#include <hip/hip_runtime.h>
#include <math.h>
#include <stdint.h>

// Problem constants (fixed by the reference):
//   B=64 batches, T=1024 time steps, C=256 classes (blank = C-1),
//   L=64 labels -> S = 2L+1 = 129 extended states.
#define B_     64
#define T_     1024
#define C_     256
#define L_     64
#define S_     (2 * L_ + 1)     // 129
#define BLANK_ (C_ - 1)         // 255
#define EPS_   1e-7f
#define NEG_   (-1e30f)
#define NTHR_  160              // 5 wave32: 129 DP states + staging lanes
#define APAD_  2                // NEG guard slots in front of alpha

// 3-way logaddexp: the three v_exp_f32 are independent (shorter critical
// path than nested 2-way). NEG_-safe: all-NEG -> m + ln3 ~= NEG.
__device__ __forceinline__ float lae3(float a, float b, float c) {
    float m = fmaxf(fmaxf(a, b), c);
    float t = __expf(a - m) + __expf(b - m) + __expf(c - m);
    return m + __logf(t);
}
__device__ __forceinline__ float lae(float a, float b) {
    float m = fmaxf(a, b);
    return m + __logf(1.0f + __expf(-fabsf(a - b)));
}

// addrspace(3) offset of an LDS object: flat LDS addresses are
// {SHARED_BASE[63:32], lds_offset[31:0]} on gfx1250, so truncation works.
__device__ __forceinline__ uint32_t lds_off(const void* p) {
    return (uint32_t)(uintptr_t)p;
}

__global__ __launch_bounds__(NTHR_)
void ctc_loss_kernel(const int*  __restrict__ y_true,   // (B, L)
                     const float* __restrict__ y_pred,  // (B, T, C) probabilities
                     float*       __restrict__ out)     // (B, 1)
{
    // Triple-buffered emission rows (1 KB each), 16B aligned for B128 async.
    __shared__ __align__(16) float rowbuf[3][C_];
    // Double-buffered alpha with 2 NEG guard slots in front: unconditional
    // A[s-1], A[s-2] reads need no boundary selects.
    __shared__ float abuf[2][S_ + APAD_ + 1];

    const int b   = blockIdx.x;
    const int tid = threadIdx.x;
    const int s   = tid;                    // extended-label state owned by this thread

    // ---- extended label & skip-transition flag for state s ----------------
    int  ext_s = BLANK_;
    bool skip  = false;                     // false for s<3 (matches reference)
    if (s < S_ && (s & 1)) {
        ext_s = y_true[b * L_ + ((s - 1) >> 1)];
        if (s >= 3) {
            int prev = y_true[b * L_ + ((s - 3) >> 1)];
            skip = (ext_s != prev);
        }
    }

    const float* base = y_pred + (size_t)b * T_ * C_;

    // ---- alpha0 from row 0 (only states 0 and 1 are reachable) ------------
    if (s < S_) {
        float a = NEG_;
        if (s == 0) a = __logf(base[BLANK_] + EPS_);
        if (s == 1) a = __logf(base[ext_s]  + EPS_);
        abuf[0][APAD_ + s] = a;
    }
    if (tid == 0) {     // guard slots, written once, never overwritten
        abuf[0][0] = NEG_; abuf[0][1] = NEG_;
        abuf[1][0] = NEG_; abuf[1][1] = NEG_;
    }

    // Staging lanes: tid 0..63 (waves 0-1) each copy 16 bytes of a row, so a
    // full 1 KB row is two global_load_async_to_lds_b128 instructions.
    const bool stager = (tid < 64);
    const uint32_t lr0 = lds_off(&rowbuf[0][tid * 4]);
    const uint32_t lr1 = lds_off(&rowbuf[1][tid * 4]);
    const uint32_t lr2 = lds_off(&rowbuf[2][tid * 4]);

    // Strength-reduced async source address (this lane's 16B slice of row
    // t+2, starting at t=1) and deep-prefetch pointer (row t+8).
    uint64_t     gg = (uint64_t)(uintptr_t)(base + (size_t)3 * C_ + tid * 4);
    const float* pf = base + (size_t)9 * C_ + tid * 4;

    // ---- preload rows 1 and 2 ---------------------------------------------
    if (stager) {
        uint64_t g1 = (uint64_t)(uintptr_t)(base + (size_t)1 * C_ + tid * 4);
        asm volatile("global_load_async_to_lds_b128 %0, %1, off"
                     :: "v"(lr1), "v"(g1) : "memory");
        uint64_t g2 = (uint64_t)(uintptr_t)(base + (size_t)2 * C_ + tid * 4);
        asm volatile("global_load_async_to_lds_b128 %0, %1, off"
                     :: "v"(lr2), "v"(g2) : "memory");
    }
    // Row 1 must be resident (<=1 outstanding: row 2 may still be in flight).
    asm volatile("s_wait_asynccnt 0x1" ::: "memory");
    __syncthreads();   // rowbuf[1] + abuf[0] + guards visible to all waves

    // ---- forward recursion, steps t = 1 .. T-1 (1023 = 6*169 + 9) ---------
    // Main loop unrolled by 6: rowbuf index AND alpha parity are compile-time
    // constants, and every guard is provably true (t <= 1014), so the body is
    // branch-free: issue row t+2, prefetch row t+8, DP update, wait, barrier.
#define STEP_FAST(PO, CURBUF, LRNXT)                                          \
    {                                                                         \
        if (stager) {                                                         \
            asm volatile("global_load_async_to_lds_b128 %0, %1, off"          \
                         :: "v"(LRNXT), "v"(gg) : "memory");                  \
            __builtin_prefetch(pf, 0, 1);                                     \
            gg += (uint64_t)(C_ * 4);                                         \
            pf += C_;                                                         \
        }                                                                     \
        if (s < S_) {                                                         \
            const float* Ap = &abuf[PO][APAD_];                               \
            float a0 = Ap[s];                                                 \
            float a1 = Ap[s - 1];               /* guard slot if s==0 */      \
            float a2 = skip ? Ap[s - 2] : NEG_; /* skip==false for s<3 */     \
            float e  = __logf((CURBUF)[ext_s] + EPS_);                        \
            abuf[(PO) ^ 1][APAD_ + s] = lae3(a0, a1, a2) + e;                 \
        }                                                                     \
        asm volatile("s_wait_asynccnt 0x1" ::: "memory");                     \
        __syncthreads();                                                      \
    }

    for (int it = 0; it < 169; ++it) {      // t = 1+6*it .. 6+6*it <= 1014
        STEP_FAST(0, rowbuf[1], lr0)
        STEP_FAST(1, rowbuf[2], lr1)
        STEP_FAST(0, rowbuf[0], lr2)
        STEP_FAST(1, rowbuf[1], lr0)
        STEP_FAST(0, rowbuf[2], lr1)
        STEP_FAST(1, rowbuf[0], lr2)
    }
#undef STEP_FAST

    // Guarded tail: t = 1015 .. 1023 (9 steps, po re-enters at 0).
    int po = 0;
    for (int t = 1015; t < T_; ++t) {
        const int cur = t % 3;
        if (stager && (t + 2 < T_)) {
            uint64_t g = (uint64_t)(uintptr_t)(base + (size_t)(t + 2) * C_ + tid * 4);
            uint32_t lrn = (cur == 1) ? lr0 : (cur == 2 ? lr1 : lr2);
            asm volatile("global_load_async_to_lds_b128 %0, %1, off"
                         :: "v"(lrn), "v"(g) : "memory");
        }
        const int pn = po ^ 1;
        if (s < S_) {
            const float* Ap = &abuf[po][APAD_];
            float a0 = Ap[s];
            float a1 = Ap[s - 1];
            float a2 = skip ? Ap[s - 2] : NEG_;
            float e  = __logf(rowbuf[cur][ext_s] + EPS_);
            abuf[pn][APAD_ + s] = lae3(a0, a1, a2) + e;
        }
        po = pn;
        // Loads retire in order: after the last issue (t=1021) the only
        // possible in-flight copy IS row t+1, so drain with 0x0.
        if (t + 2 < T_) {
            asm volatile("s_wait_asynccnt 0x1" ::: "memory");
        } else {
            asm volatile("s_wait_asynccnt 0x0" ::: "memory");
        }
        __syncthreads();
    }

    // ---- loss: end in final blank or final label --------------------------
    if (s == 0) {
        float ll = lae(abuf[po][APAD_ + S_ - 1], abuf[po][APAD_ + S_ - 2]);
        out[b] = -ll;
    }
}

extern "C" void kernel_launch(void* const* d_in, const int* in_sizes, int n_in,
                              void* d_out, int out_size, void* d_ws, size_t ws_size,
                              hipStream_t stream) {
    const int*   y_true = (const int*)  d_in[0];   // (B, L) int32
    const float* y_pred = (const float*)d_in[1];   // (B, T, C) float32
    float*       out    = (float*)      d_out;     // (B, 1) float32

    (void)in_sizes; (void)n_in; (void)d_ws; (void)ws_size; (void)out_size;

    hipLaunchKernelGGL(ctc_loss_kernel, dim3(B_), dim3(NTHR_), 0, stream,
                       y_true, y_pred, out);
}